// DiffusionLayer_88270167868014
// MI455X (gfx1250) — compile-verified
//
#include <hip/hip_runtime.h>

typedef __attribute__((ext_vector_type(2))) float v2f;
typedef __attribute__((ext_vector_type(8))) float v8f;

#define S 32
#define EPSF 1e-6f
#define TS_H 529   // per-h plane stride in LDS words (odd -> conflict-free lane spread)
#define MSTR 33    // per-tile stride within a plane (odd)

// ---------------------------------------------------------------------------
// Kernel 1: build the 11 per-sweep inverse (or squared-inverse) 32x32 matrices
// for every (sweep, channel, row) and store them pre-swizzled in the WMMA
// B-fragment layout:  Bbuf[mi*1024 + half*512 + kk*64 + lane*2 + v]
//                       = M[n = half*16 + lane%16][k = 4*kk + v + 2*(lane/16)]
// Sweep schedule (merged): s even = x-type, s odd = y-type.
//   x: t = 0.1*(s/2), dt_eff = DT/2 = 0.05, squared for 0 < s < 10
//   y: t = 0.05 + 0.1*(s/2), dt_eff = DT = 0.1
// ---------------------------------------------------------------------------
__global__ __launch_bounds__(32) void precompute_mats(
    const float* __restrict__ abase, const float* __restrict__ bbase,
    const float* __restrict__ atc,   const float* __restrict__ btc,
    float* __restrict__ Bbuf)
{
    const int mi   = blockIdx.x;          // 0..1055
    const int s    = mi / (3 * S);
    const int rem  = mi % (3 * S);
    const int c    = rem / S;
    const int r    = rem % S;             // row index (h for x-type, w for y-type)
    const int lane = threadIdx.x;

    const bool isX = ((s & 1) == 0);
    float t, dteff; bool sq = false;
    if (isX) { t = 0.1f * (float)(s >> 1); dteff = 0.05f; sq = (s > 0 && s < 10); }
    else     { t = 0.05f + 0.1f * (float)(s >> 1); dteff = 0.1f; }

    const float* base = isX ? abase : bbase;
    const float* tc   = isX ? atc   : btc;

    // clipped coefficient along the solve axis (all lanes compute identically)
    float cv[S];
    #pragma unroll
    for (int k = 0; k < S; k++) {
        int h = isX ? r : k;
        int w = isX ? k : r;
        int idx = (c * S + h) * S + w;
        cv[k] = fmaxf(base[idx] + tc[idx] * t, EPSF);
    }
    // 3-tap replicate-pad mean, scaled by dt/(dh*dh) with dh = 1
    float cs[S];
    #pragma unroll
    for (int k = 0; k < S; k++) {
        float vm = cv[(k == 0) ? 0 : k - 1];
        float vp = cv[(k == S - 1) ? S - 1 : k + 1];
        cs[k] = ((vm + cv[k] + vp) * (1.0f / 3.0f)) * dteff;
    }

    // Thomas solve for RHS = e_lane  -> column `lane` of T^{-1}
    float cst[S], dst[S];
    float cprev = 0.0f, dprev = 0.0f;
    #pragma unroll
    for (int k = 0; k < S; k++) {
        float bk = 1.0f + 2.0f * cs[k];
        if (k == 0)     bk = 1.0f + cs[0];
        if (k == S - 1) bk = 1.0f + cs[S - 1];
        float ak = (k == 0) ? 0.0f : -cs[k];
        float ck = -cs[k];
        float dk = (k == lane) ? 1.0f : 0.0f;
        float denom = bk - ak * cprev + EPSF;
        float inv = 1.0f / denom;
        cst[k] = ck * inv;
        dst[k] = (dk - ak * dprev) * inv;
        cprev = cst[k]; dprev = dst[k];
    }
    float x[S];
    x[S - 1] = dst[S - 1];
    #pragma unroll
    for (int k = S - 2; k >= 0; k--) x[k] = dst[k] - cst[k] * x[k + 1];

    __shared__ float Ml[S * 33];          // M[i][j] at Ml[i*33 + j]
    #pragma unroll
    for (int i = 0; i < S; i++) Ml[i * 33 + lane] = x[i];
    __syncthreads();

    if (sq) {                              // M2 column j = M * (column j of M)
        float y[S];
        #pragma unroll
        for (int i = 0; i < S; i++) {
            float acc = 0.0f;
            #pragma unroll
            for (int k = 0; k < S; k++) acc += Ml[i * 33 + k] * x[k];
            y[i] = acc;
        }
        __syncthreads();
        #pragma unroll
        for (int i = 0; i < S; i++) Ml[i * 33 + lane] = y[i];
        __syncthreads();
    }

    // Emit in WMMA B-fragment order (b64-loadable by the main kernel)
    const size_t obase = (size_t)mi * 1024;
    const int g  = lane >> 4;
    const int nl = lane & 15;
    #pragma unroll
    for (int half = 0; half < 2; half++) {
        #pragma unroll
        for (int kk = 0; kk < 8; kk++) {
            #pragma unroll
            for (int v = 0; v < 2; v++) {
                int n = half * 16 + nl;
                int k = 4 * kk + v + 2 * g;
                Bbuf[obase + half * 512 + kk * 64 + lane * 2 + v] = Ml[n * 33 + k];
            }
        }
    }
}

// ---------------------------------------------------------------------------
// Sweep bodies, fully specialized at compile time (no runtime x/y branching
// in the inner loops). Each wave handles 4 rows; rows are independent, so no
// intra-sweep synchronization is needed.
// ---------------------------------------------------------------------------
__device__ __forceinline__ void do_rows(
    float* lds, const float* __restrict__ Bbuf,
    int sweep_c_base, int wv, int lane, bool is_x)
{
    const int g  = lane >> 4;
    const int nl = lane & 15;

    #pragma unroll 1
    for (int i = 0; i < 4; i++) {
        const int r = wv * 4 + i;         // solve row (h for x-sweep, w for y-sweep)

        // B fragments (L2-hot, pre-swizzled): two N-halves, 8 K-chunks
        const float* bp = Bbuf + (size_t)(sweep_c_base + r) * 1024 + lane * 2;
        v2f bf0[8], bf1[8];
        #pragma unroll
        for (int kk = 0; kk < 8; kk++) {
            bf0[kk] = *(const v2f*)(bp + kk * 64);
            bf1[kk] = *(const v2f*)(bp + 512 + kk * 64);
        }

        // A fragments from LDS: A[m][k] layout (lane = m + 16*(k>=2), vgpr = k&1)
        v2f af[8];
        if (is_x) {
            const int ab = r * TS_H + nl * MSTR + 2 * g;
            #pragma unroll
            for (int kk = 0; kk < 8; kk++) {
                v2f t2; t2.x = lds[ab + 4 * kk]; t2.y = lds[ab + 4 * kk + 1];
                af[kk] = t2;
            }
        } else {
            const int ab = 2 * g * TS_H + nl * MSTR + r;
            #pragma unroll
            for (int kk = 0; kk < 8; kk++) {
                v2f t2; t2.x = lds[ab + 4 * kk * TS_H]; t2.y = lds[ab + (4 * kk + 1) * TS_H];
                af[kk] = t2;
            }
        }

        v8f c0 = {}; v8f c1 = {};
        #pragma unroll
        for (int kk = 0; kk < 8; kk++) {
            c0 = __builtin_amdgcn_wmma_f32_16x16x4_f32(
                    false, af[kk], false, bf0[kk], (short)0, c0, false, false);
            c1 = __builtin_amdgcn_wmma_f32_16x16x4_f32(
                    false, af[kk], false, bf1[kk], (short)0, c1, false, false);
        }

        // D[m][n]: vgpr rr, lanes 0-15 -> m=rr, lanes 16-31 -> m=rr+8
        if (is_x) {
            const int db = r * TS_H + 8 * g * MSTR + nl;
            #pragma unroll
            for (int rr = 0; rr < 8; rr++) {
                lds[db + rr * MSTR]      = c0[rr];
                lds[db + rr * MSTR + 16] = c1[rr];
            }
        } else {
            const int db = nl * TS_H + 8 * g * MSTR + r;
            #pragma unroll
            for (int rr = 0; rr < 8; rr++) {
                lds[db + rr * MSTR]              = c0[rr];
                lds[db + rr * MSTR + 16 * TS_H]  = c1[rr];
            }
        }
    }
}

// ---------------------------------------------------------------------------
// Kernel 2: fully fused ADI. One workgroup = 16 batch-tiles of one channel,
// resident in LDS across all 11 sweeps. Each sweep row/column solve is
// Out[16b x 32] = In[16b x 32] * M^T via v_wmma_f32_16x16x4_f32.
// LDS layout: lds[h*TS_H + m*MSTR + w], odd strides for bank spread.
// ---------------------------------------------------------------------------
__global__ __launch_bounds__(256) void diffuse_fused(
    const float* __restrict__ u, const float* __restrict__ Bbuf,
    float* __restrict__ out)
{
    __shared__ float lds[S * TS_H];       // 32*529*4 = 67712 B (<< 320KB/WGP)

    const int tid  = threadIdx.x;
    const int lane = tid & 31;
    const int wv   = tid >> 5;            // wave id 0..7
    const int c    = blockIdx.x % 3;
    const int bg   = blockIdx.x / 3;

    // ---- load 16 tiles (each 32x32 fp32, contiguous in global) ----
    #pragma unroll 1
    for (int m = 0; m < 16; m++) {
        size_t gb = ((size_t)(bg * 16 + m) * 3 + c) * 1024;
        float4 v = *(const float4*)(u + gb + tid * 4);
        int h = tid >> 3;
        int w = (tid & 7) << 2;
        int a = h * TS_H + m * MSTR + w;
        lds[a] = v.x; lds[a + 1] = v.y; lds[a + 2] = v.z; lds[a + 3] = v.w;
    }
    __syncthreads();

    // sweeps: x(0) y(1) x(2) y(3) ... x(8) y(9) x(10)
    #pragma unroll 1
    for (int sp = 0; sp < 6; sp++) {
        const int sx = 2 * sp;
        do_rows(lds, Bbuf, (sx * 3 + c) * S, wv, lane, true);   // x-type
        __syncthreads();
        if (sp < 5) {
            do_rows(lds, Bbuf, ((sx + 1) * 3 + c) * S, wv, lane, false);  // y-type
            __syncthreads();
        }
    }

    // ---- write tiles back ----
    #pragma unroll 1
    for (int m = 0; m < 16; m++) {
        int h = tid >> 3;
        int w = (tid & 7) << 2;
        int a = h * TS_H + m * MSTR + w;
        float4 v;
        v.x = lds[a]; v.y = lds[a + 1]; v.z = lds[a + 2]; v.w = lds[a + 3];
        size_t gb = ((size_t)(bg * 16 + m) * 3 + c) * 1024;
        *(float4*)(out + gb + tid * 4) = v;
    }
}

// ---------------------------------------------------------------------------
extern "C" void kernel_launch(void* const* d_in, const int* in_sizes, int n_in,
                              void* d_out, int out_size, void* d_ws, size_t ws_size,
                              hipStream_t stream) {
    const float* u     = (const float*)d_in[0];
    const float* abase = (const float*)d_in[1];
    const float* bbase = (const float*)d_in[2];
    const float* atc   = (const float*)d_in[3];
    const float* btc   = (const float*)d_in[4];
    float* outp = (float*)d_out;
    float* Bbuf = (float*)d_ws;           // needs 11*3*32*1024*4 = 4.3 MB

    const int nB = in_sizes[0] / (3 * S * S);   // 2048

    precompute_mats<<<11 * 3 * S, 32, 0, stream>>>(abase, bbase, atc, btc, Bbuf);
    diffuse_fused<<<(nB / 16) * 3, 256, 0, stream>>>(u, Bbuf, outp);
}